// FullModel_65352222376343
// MI455X (gfx1250) — compile-verified
//
#include <hip/hip_runtime.h>

// Model constants (match reference)
#define BSZ   4096
#define CCH   4
#define KK    64
#define TT    64
#define NL2   8

typedef __attribute__((ext_vector_type(2))) float v2f;
typedef __attribute__((ext_vector_type(8))) float v8f;

__device__ __forceinline__ v8f v8zero() {
  v8f z;
  for (int i = 0; i < 8; ++i) z[i] = 0.f;
  return z;
}

// ---------------------------------------------------------------------------
// Kernel 1: matcher.  block = 64 batch x 4 k-groups, one channel per blockIdx.y
// x1[b,c,k] = softargmax_t( leaky-scan of w[c,k,t]*exp((arr - tref)/40) )
// ---------------------------------------------------------------------------
__global__ __launch_bounds__(256) void matcher_kernel(
    const float* __restrict__ lat,    // (B,C,T)
    const float* __restrict__ mw,     // (C,K,T)
    const float* __restrict__ mdel,   // (C,K,T)
    float* __restrict__ x1)           // (B,C,K)
{
  const int c  = blockIdx.y;
  const int b0 = blockIdx.x * 64;
  __shared__ float st[64][65];
  for (int idx = threadIdx.x; idx < 64 * 64; idx += 256) {
    int bl = idx >> 6, t = idx & 63;
    st[bl][t] = lat[((size_t)(b0 + bl) * CCH + c) * TT + t];
  }
  __syncthreads();

  const int bl = threadIdx.x & 63;
  const int kg = threadIdx.x >> 6;
  for (int k = kg * 16; k < kg * 16 + 16; ++k) {
    const float* dk = mdel + ((size_t)c * KK + k) * TT;
    const float* wk = mw   + ((size_t)c * KK + k) * TT;
    __builtin_prefetch(dk, 0, 3);
    __builtin_prefetch(wk, 0, 3);
    // pass 1: t_ref = max arrival
    float tref = -1e30f;
    for (int t = 0; t < TT; ++t) {
      float arr = st[bl][t] + dk[t];
      tref = fmaxf(tref, arr);
    }
    // pass 2: fused I, leaky scan (dt/tau_mem = 0.05), online soft-argmax (tau=1)
    float v = 0.f, num = 0.f, den = 0.f;
    for (int t = 0; t < TT; ++t) {
      float arr = st[bl][t] + dk[t];
      float I = wk[t] * __expf((arr - tref) * 0.025f);   // 1/TAU_PSP = 1/40
      v += (I - v) * 0.05f;
      float e = __expf(v);
      den += e;
      num += (float)t * e;
    }
    x1[((size_t)(b0 + bl) * CCH + c) * KK + k] = num / den;
  }
}

// ---------------------------------------------------------------------------
// Kernel 2: dendritic layer 1 (64 -> 32, 50 steps) via V_WMMA_F32_16X16X4_F32.
// One wave per 16-batch tile per channel. psp(t) = [t>=st]*exp(st/5)*exp(-t/5).
// conn mask: output o <- inputs {2o, 2o+1}  => out-tile j only uses K-chunks [8j,8j+8)
// ---------------------------------------------------------------------------
__global__ __launch_bounds__(128) void layer1_kernel(
    const float* __restrict__ x1,   // (B,C,64)
    const float* __restrict__ W1,   // (C,32,64)
    float* __restrict__ x2)         // (B,C,32)
{
  const int c    = blockIdx.y;
  const int lane = threadIdx.x & 31;
  const int wave = threadIdx.x >> 5;
  const int tile = blockIdx.x * 4 + wave;     // 0..255
  const int m    = lane & 15;                 // A/D row within tile (also N col)
  const int hi   = lane >> 4;                 // 0: K {0,1}, 1: K {2,3} within chunk
  const int o0   = hi * 2;
  const int b    = tile * 16 + m;

  // spike times + hoisted exp(st/5) for this lane's 32 assigned inputs
  float stv[32], ev[32];
  const float* xrow = x1 + ((size_t)b * CCH + c) * 64;
  for (int kc = 0; kc < 16; ++kc) {
    float s0 = xrow[4 * kc + o0];
    float s1 = xrow[4 * kc + o0 + 1];
    stv[2 * kc]     = s0;  ev[2 * kc]     = __expf(s0 * 0.2f);
    stv[2 * kc + 1] = s1;  ev[2 * kc + 1] = __expf(s1 * 0.2f);
  }

  // B fragments: Weff^T = (W1 * conn)^T, resident in registers
  const int n = m;
  v2f Bf[2][8];
  for (int j = 0; j < 2; ++j) {
    int o = j * 16 + n;
    const float* wrow = W1 + ((size_t)c * 32 + o) * 64;
    for (int kk = 0; kk < 8; ++kk) {
      int i0 = 4 * (j * 8 + kk) + o0;
      v2f bb;
      bb.x = ((i0 >> 1) == o)       ? wrow[i0]     : 0.f;
      bb.y = (((i0 + 1) >> 1) == o) ? wrow[i0 + 1] : 0.f;
      Bf[j][kk] = bb;
    }
  }

  v8f vst0 = v8zero(), vst1 = v8zero();
  v8f num0 = v8zero(), num1 = v8zero();
  v8f den0 = v8zero(), den1 = v8zero();

  for (int t = 0; t < 50; ++t) {
    float tf = (float)t;
    float ex = __expf(-tf * 0.2f);            // exp(-t/5)
    v8f acc[2];
    for (int j = 0; j < 2; ++j) {
      v8f a = v8zero();
      for (int kk = 0; kk < 8; ++kk) {
        int kc = j * 8 + kk;
        v2f A;
        A.x = (tf >= stv[2 * kc])     ? ev[2 * kc]     * ex : 0.f;
        A.y = (tf >= stv[2 * kc + 1]) ? ev[2 * kc + 1] * ex : 0.f;
        a = __builtin_amdgcn_wmma_f32_16x16x4_f32(false, A, false, Bf[j][kk],
                                                  (short)0, a, false, false);
      }
      acc[j] = a;
    }
    for (int r = 0; r < 8; ++r) {
      float v0n = vst0[r] + (acc[0][r] - vst0[r]) * 0.05f;
      vst0[r] = v0n;
      float e0 = __expf(v0n);
      den0[r] += e0; num0[r] += tf * e0;
      float v1n = vst1[r] + (acc[1][r] - vst1[r]) * 0.05f;
      vst1[r] = v1n;
      float e1 = __expf(v1n);
      den1[r] += e1; num1[r] += tf * e1;
    }
  }
  for (int r = 0; r < 8; ++r) {
    int bb = tile * 16 + r + hi * 8;          // C/D layout: M = r (+8 for hi lanes)
    x2[((size_t)bb * CCH + c) * 32 + n]      = num0[r] / den0[r];
    x2[((size_t)bb * CCH + c) * 32 + 16 + n] = num1[r] / den1[r];
  }
}

// ---------------------------------------------------------------------------
// Kernel 3: dendritic layer 2 (32 -> 8, 60 steps) via WMMA f32 16x16x4.
// conn: output o <- inputs [4o, 4o+4).  Single 16x16 output tile, N<8 valid.
// ---------------------------------------------------------------------------
__global__ __launch_bounds__(128) void layer2_kernel(
    const float* __restrict__ x2,   // (B,C,32)
    const float* __restrict__ W2,   // (C,8,32)
    float* __restrict__ x3)         // (B,C,8)
{
  const int c    = blockIdx.y;
  const int lane = threadIdx.x & 31;
  const int wave = threadIdx.x >> 5;
  const int tile = blockIdx.x * 4 + wave;
  const int m    = lane & 15;
  const int hi   = lane >> 4;
  const int o0   = hi * 2;
  const int b    = tile * 16 + m;

  float stv[16], ev[16];
  const float* xrow = x2 + ((size_t)b * CCH + c) * 32;
  for (int kc = 0; kc < 8; ++kc) {
    float s0 = xrow[4 * kc + o0];
    float s1 = xrow[4 * kc + o0 + 1];
    stv[2 * kc]     = s0;  ev[2 * kc]     = __expf(s0 * 0.2f);
    stv[2 * kc + 1] = s1;  ev[2 * kc + 1] = __expf(s1 * 0.2f);
  }

  const int n = m;                            // output index (valid when n < 8)
  v2f Bf[8];
  for (int kk = 0; kk < 8; ++kk) {
    int i0 = 4 * kk + o0;
    v2f bb; bb.x = 0.f; bb.y = 0.f;
    if (n < 8) {
      const float* wrow = W2 + ((size_t)c * 8 + n) * 32;
      if ((i0 >> 2) == n)       bb.x = wrow[i0];
      if (((i0 + 1) >> 2) == n) bb.y = wrow[i0 + 1];
    }
    Bf[kk] = bb;
  }

  v8f vst = v8zero(), num = v8zero(), den = v8zero();
  for (int t = 0; t < 60; ++t) {
    float tf = (float)t;
    float ex = __expf(-tf * 0.2f);
    v8f a = v8zero();
    for (int kk = 0; kk < 8; ++kk) {
      v2f A;
      A.x = (tf >= stv[2 * kk])     ? ev[2 * kk]     * ex : 0.f;
      A.y = (tf >= stv[2 * kk + 1]) ? ev[2 * kk + 1] * ex : 0.f;
      a = __builtin_amdgcn_wmma_f32_16x16x4_f32(false, A, false, Bf[kk],
                                                (short)0, a, false, false);
    }
    for (int r = 0; r < 8; ++r) {
      float vn = vst[r] + (a[r] - vst[r]) * 0.05f;
      vst[r] = vn;
      float e = __expf(vn);
      den[r] += e; num[r] += tf * e;
    }
  }
  if (n < 8) {
    for (int r = 0; r < 8; ++r) {
      int bb = tile * 16 + r + hi * 8;
      x3[((size_t)bb * CCH + c) * 8 + n] = num[r] / den[r];
    }
  }
}

// ---------------------------------------------------------------------------
// Kernel 4: soma layer (8 -> 1, 75 steps) + cross-channel recurrence (80 steps)
// + gates output.  One thread per batch item.
// ---------------------------------------------------------------------------
__global__ __launch_bounds__(256) void final_kernel(
    const float* __restrict__ x3,    // (B,C,8)
    const float* __restrict__ W3,    // (C,1,8)
    const float* __restrict__ gl,    // (C,C)
    float* __restrict__ out_recon,   // (B,C)
    float* __restrict__ out_gates,   // (C,C)
    float* __restrict__ out_soma)    // (B,C)
{
  const int b = blockIdx.x * 256 + threadIdx.x;

  // gates = sigmoid(gate_logits) * (1 - eye)
  float gates[4][4];
  for (int r = 0; r < 4; ++r)
    for (int d = 0; d < 4; ++d)
      gates[r][d] = (r == d) ? 0.f : 1.f / (1.f + __expf(-gl[r * 4 + d]));
  if (blockIdx.x == 0 && threadIdx.x < 16)
    out_gates[threadIdx.x] = gates[threadIdx.x >> 2][threadIdx.x & 3];

  if (b >= BSZ) return;

  // soma layer per channel: psp tau 5, mem tau 20, 75 steps, dense 8->1
  float soma[4];
  for (int c = 0; c < 4; ++c) {
    float st[8], wE[8];
    for (int i = 0; i < 8; ++i) {
      float s = x3[((size_t)b * CCH + c) * 8 + i];
      st[i] = s;
      wE[i] = W3[c * 8 + i] * __expf(s * 0.2f);
    }
    float v = 0.f, num = 0.f, den = 0.f;
    for (int t = 0; t < 75; ++t) {
      float tf = (float)t;
      float ex = __expf(-tf * 0.2f);
      float I = 0.f;
      for (int i = 0; i < 8; ++i)
        I += (tf >= st[i]) ? wE[i] * ex : 0.f;
      v += (I - v) * 0.05f;
      float e = __expf(v);
      den += e; num += tf * e;
    }
    soma[c] = num / den;
    out_soma[(size_t)b * CCH + c] = soma[c];
  }

  // cross-channel: psp tau 10, mem tau 20, 80 steps
  float Es[4], vv[4], num[4], den[4];
  for (int c = 0; c < 4; ++c) {
    Es[c] = __expf(soma[c] * 0.1f);
    vv[c] = 0.f; num[c] = 0.f; den[c] = 0.f;
  }
  for (int t = 0; t < 80; ++t) {
    float tf = (float)t;
    float ex = __expf(-tf * 0.1f);
    float psp[4];
    for (int d = 0; d < 4; ++d)
      psp[d] = (tf >= soma[d]) ? Es[d] * ex : 0.f;
    for (int c = 0; c < 4; ++c) {
      float I = 0.f;
      for (int d = 0; d < 4; ++d) I += gates[c][d] * psp[d];
      vv[c] += (I - vv[c]) * 0.05f;
      float e = __expf(vv[c]);
      den[c] += e; num[c] += tf * e;
    }
  }
  for (int c = 0; c < 4; ++c)
    out_recon[(size_t)b * CCH + c] = num[c] / den[c];
}

// ---------------------------------------------------------------------------
extern "C" void kernel_launch(void* const* d_in, const int* in_sizes, int n_in,
                              void* d_out, int out_size, void* d_ws, size_t ws_size,
                              hipStream_t stream) {
  const float* lat  = (const float*)d_in[0];   // latencies (B,C,T)
  const float* mw   = (const float*)d_in[1];   // matcher_w (C,K,T)
  const float* mdel = (const float*)d_in[2];   // matcher_delays (C,K,T)
  const float* W1   = (const float*)d_in[3];   // (C,32,64)
  const float* W2   = (const float*)d_in[4];   // (C,8,32)
  const float* W3   = (const float*)d_in[5];   // (C,1,8)
  const float* gl   = (const float*)d_in[6];   // (C,C)

  float* out   = (float*)d_out;
  float* recon = out;                          // (B,C)
  float* gates = out + (size_t)BSZ * CCH;      // (C,C)
  float* soma  = gates + CCH * CCH;            // (B,C)

  float* x1 = (float*)d_ws;                    // (B,C,64)
  float* x2 = x1 + (size_t)BSZ * CCH * KK;     // (B,C,32)
  float* x3 = x2 + (size_t)BSZ * CCH * 32;     // (B,C,8)

  matcher_kernel<<<dim3(BSZ / 64, CCH), 256, 0, stream>>>(lat, mw, mdel, x1);
  layer1_kernel<<<dim3(BSZ / 64, CCH), 128, 0, stream>>>(x1, W1, x2);
  layer2_kernel<<<dim3(BSZ / 64, CCH), 128, 0, stream>>>(x2, W2, x3);
  final_kernel<<<dim3(BSZ / 256), 256, 0, stream>>>(x3, W3, gl, recon, gates, soma);
}